// EGNNLayer_18837726560908
// MI455X (gfx1250) — compile-verified
//
#include <hip/hip_runtime.h>

#define N_NODES   10000
#define N_EDGES   640000
#define D         128
#define T         64
#define K_EDGE    321      // 2*D + 1 + T
#define K_EDGE_PAD 324     // padded to multiple of 4
#define K_NODE    320      // 2*D + T (already multiple of 4)

typedef __attribute__((ext_vector_type(2))) float v2f;
typedef __attribute__((ext_vector_type(8))) float v8f;

// D = A(16x4) * B(4x16) + C, full fp32 (reference precision).
__device__ __forceinline__ v8f wmma4(v2f a, v2f b, v8f c) {
    return __builtin_amdgcn_wmma_f32_16x16x4_f32(
        /*neg_a=*/false, a, /*neg_b=*/false, b,
        /*c_mod=*/(short)0, c, /*reuse_a=*/false, /*reuse_b=*/false);
}

__device__ __forceinline__ float silu_f(float x) {
    return x / (1.0f + __expf(-x));
}

__device__ __forceinline__ v8f bias_fill(float b) {
    v8f r;
#pragma unroll
    for (int i = 0; i < 8; ++i) r[i] = b;
    return r;
}

// Generic 16x16 x Kpad tile GEMM: A rows in LDS (stride astride), B transposed
// N-major in global (Wt[n*kpad + k], zero-padded), acc pre-loaded with bias.
__device__ __forceinline__ void gemm_tile(const float* __restrict__ arow,
                                          const float* __restrict__ Wt,
                                          int kpad, int kl, int nloc,
                                          v8f acc[8]) {
    for (int k0 = 0; k0 < kpad; k0 += 4) {
        int kr = k0 + kl * 2;
        v2f a = *(const v2f*)(arow + kr);
#pragma unroll
        for (int nt = 0; nt < 8; ++nt) {
            int n = nt * 16 + nloc;
            v2f b = *(const v2f*)(Wt + n * kpad + kr);
            acc[nt] = wmma4(a, b, acc[nt]);
        }
    }
}

// -------------------------------------------------------------------------
// Weight prep: Wt[n*kpad + k] = (k < K) ? W[k*128 + n] : 0
// -------------------------------------------------------------------------
__global__ void egnn_transpose_pad(const float* __restrict__ W,
                                   float* __restrict__ Wt,
                                   int K, int Kpad) {
    int idx = blockIdx.x * blockDim.x + threadIdx.x;
    if (idx >= D * Kpad) return;
    int n = idx / Kpad;
    int k = idx - n * Kpad;
    Wt[idx] = (k < K) ? W[k * D + n] : 0.0f;
}

// -------------------------------------------------------------------------
// Edge kernel: 2 waves / block, 32 edges / block.
// Each wave owns a 16-edge M-tile; iterates all 8 N-tiles of the 128 outputs.
// -------------------------------------------------------------------------
__global__ __launch_bounds__(64) void egnn_edge_kernel(
    const float* __restrict__ h,
    const float* __restrict__ diff_cart,
    const float* __restrict__ dist_sq,
    const int*   __restrict__ edge_src,
    const int*   __restrict__ edge_dst,
    const float* __restrict__ t_emb_edges,
    const float* __restrict__ eW1t, const float* __restrict__ eb1,
    const float* __restrict__ eW2t, const float* __restrict__ eb2,
    const float* __restrict__ cW1t, const float* __restrict__ cb1,
    const float* __restrict__ cW2,
    float* __restrict__ m_i,        // [N_NODES, D] accumulator (pre-zeroed)
    float* __restrict__ coord_out)  // [N_NODES, 3] accumulator (pre-zeroed)
{
    __shared__ float s_in[32 * K_EDGE_PAD];  // edge inputs; later reused for c1 (32*D)
    __shared__ float s_m [32 * D];           // m1, then m2 (= final m_ij)

    const int tid  = threadIdx.x;
    const int lane = tid & 31;
    const int wave = tid >> 5;            // 0..1
    const int e0   = blockIdx.x * 32;

    // ---- stage gathered edge inputs into LDS (zero-pad cols 321..323) ----
    for (int idx = tid; idx < 32 * K_EDGE_PAD; idx += 64) {
        int e  = idx / K_EDGE_PAD;
        int c  = idx - e * K_EDGE_PAD;
        int ge = e0 + e;
        float v;
        if      (c < D)        v = h[edge_src[ge] * D + c];
        else if (c < 2 * D)    v = h[edge_dst[ge] * D + (c - D)];
        else if (c == 2 * D)   v = dist_sq[ge];
        else if (c < K_EDGE)   v = t_emb_edges[ge * T + (c - (2 * D + 1))];
        else                   v = 0.0f;
        s_in[idx] = v;
    }
    __syncthreads();

    const int m    = lane & 15;           // matrix row within tile
    const int kl   = lane >> 4;           // K sub-pair selector (0/1)
    const int nloc = lane & 15;           // column within N-tile
    const int mhi  = (lane >> 4) * 8;     // C/D row offset

    v8f acc[8];

    // ---- m1 = silu(edge_in @ eW1 + eb1), K = 321 (padded 324) ----
#pragma unroll
    for (int nt = 0; nt < 8; ++nt) acc[nt] = bias_fill(eb1[nt * 16 + nloc]);
    gemm_tile(&s_in[(wave * 16 + m) * K_EDGE_PAD], eW1t, K_EDGE_PAD, kl, nloc, acc);
#pragma unroll
    for (int nt = 0; nt < 8; ++nt)
#pragma unroll
        for (int i = 0; i < 8; ++i)
            s_m[(wave * 16 + mhi + i) * D + nt * 16 + nloc] = silu_f(acc[nt][i]);
    __syncthreads();

    // ---- m2 = silu(m1 @ eW2 + eb2), K = 128 ----
    const float* mrow = &s_m[(wave * 16 + m) * D];
#pragma unroll
    for (int nt = 0; nt < 8; ++nt) acc[nt] = bias_fill(eb2[nt * 16 + nloc]);
    gemm_tile(mrow, eW2t, D, kl, nloc, acc);
    __syncthreads();   // all reads of m1 done; safe to overwrite s_m
#pragma unroll
    for (int nt = 0; nt < 8; ++nt)
#pragma unroll
        for (int i = 0; i < 8; ++i)
            s_m[(wave * 16 + mhi + i) * D + nt * 16 + nloc] = silu_f(acc[nt][i]);
    __syncthreads();   // s_m now holds final m_ij, visible to both waves

    // ---- c1 = silu(m_ij @ cW1 + cb1), K = 128; stash into s_in ----
#pragma unroll
    for (int nt = 0; nt < 8; ++nt) acc[nt] = bias_fill(cb1[nt * 16 + nloc]);
    gemm_tile(mrow, cW1t, D, kl, nloc, acc);
#pragma unroll
    for (int nt = 0; nt < 8; ++nt)
#pragma unroll
        for (int i = 0; i < 8; ++i)
            s_in[(wave * 16 + mhi + i) * D + nt * 16 + nloc] = silu_f(acc[nt][i]);
    __syncthreads();

    // ---- coord weight: per-edge 128-dot with cW2, scatter onto src nodes ----
    if (tid < 32) {
        int ge = e0 + tid;
        float sum = 0.0f;
        for (int c = 0; c < D; ++c) sum += s_in[tid * D + c] * cW2[c];
        int src = edge_src[ge];
#pragma unroll
        for (int j = 0; j < 3; ++j)
            atomicAdd(&coord_out[src * 3 + j], diff_cart[ge * 3 + j] * sum);
    }

    // ---- message scatter: m_i[dst] += m_ij ----
    for (int idx = tid; idx < 32 * D; idx += 64) {
        int e = idx >> 7;
        int c = idx & (D - 1);
        atomicAdd(&m_i[edge_dst[e0 + e] * D + c], s_m[idx]);
    }
}

// -------------------------------------------------------------------------
// Node kernel: h_out = h + (silu([h, m_i, t_emb] @ nW1 + nb1) @ nW2 + nb2)
// -------------------------------------------------------------------------
__global__ __launch_bounds__(64) void egnn_node_kernel(
    const float* __restrict__ h,
    const float* __restrict__ m_i,
    const float* __restrict__ t_emb_nodes,
    const float* __restrict__ nW1t, const float* __restrict__ nb1,
    const float* __restrict__ nW2t, const float* __restrict__ nb2,
    float* __restrict__ h_out)
{
    __shared__ float s_in[32 * K_NODE];
    __shared__ float s_x [32 * D];

    const int tid  = threadIdx.x;
    const int lane = tid & 31;
    const int wave = tid >> 5;
    const int n0   = blockIdx.x * 32;

    for (int idx = tid; idx < 32 * K_NODE; idx += 64) {
        int r  = idx / K_NODE;
        int c  = idx - r * K_NODE;
        int gn = n0 + r;
        float v = 0.0f;
        if (gn < N_NODES) {
            if      (c < D)      v = h[gn * D + c];
            else if (c < 2 * D)  v = m_i[gn * D + (c - D)];
            else                 v = t_emb_nodes[gn * T + (c - 2 * D)];
        }
        s_in[idx] = v;
    }
    __syncthreads();

    const int m    = lane & 15;
    const int kl   = lane >> 4;
    const int nloc = lane & 15;
    const int mhi  = (lane >> 4) * 8;

    v8f acc[8];

    // ---- x1 = silu(node_in @ nW1 + nb1), K = 320 ----
#pragma unroll
    for (int nt = 0; nt < 8; ++nt) acc[nt] = bias_fill(nb1[nt * 16 + nloc]);
    gemm_tile(&s_in[(wave * 16 + m) * K_NODE], nW1t, K_NODE, kl, nloc, acc);
#pragma unroll
    for (int nt = 0; nt < 8; ++nt)
#pragma unroll
        for (int i = 0; i < 8; ++i)
            s_x[(wave * 16 + mhi + i) * D + nt * 16 + nloc] = silu_f(acc[nt][i]);
    __syncthreads();

    // ---- h_out = x1 @ nW2 + nb2 + h, K = 128 ----
#pragma unroll
    for (int nt = 0; nt < 8; ++nt) acc[nt] = bias_fill(nb2[nt * 16 + nloc]);
    gemm_tile(&s_x[(wave * 16 + m) * D], nW2t, D, kl, nloc, acc);
#pragma unroll
    for (int nt = 0; nt < 8; ++nt) {
        int n = nt * 16 + nloc;
#pragma unroll
        for (int i = 0; i < 8; ++i) {
            int row = n0 + wave * 16 + mhi + i;
            if (row < N_NODES)
                h_out[row * D + n] = acc[nt][i] + h[row * D + n];
        }
    }
}

// -------------------------------------------------------------------------
__global__ void egnn_zero_kernel(float* __restrict__ m_i,
                                 float* __restrict__ coord_out)
{
    int i = blockIdx.x * blockDim.x + threadIdx.x;
    if (i < N_NODES * D) m_i[i] = 0.0f;
    if (i < N_NODES * 3) coord_out[i] = 0.0f;
}

// -------------------------------------------------------------------------
extern "C" void kernel_launch(void* const* d_in, const int* in_sizes, int n_in,
                              void* d_out, int out_size, void* d_ws, size_t ws_size,
                              hipStream_t stream) {
    const float* h       = (const float*)d_in[0];
    const float* diff    = (const float*)d_in[1];
    const float* dist_sq = (const float*)d_in[2];
    const int*   esrc    = (const int*)  d_in[3];
    const int*   edst    = (const int*)  d_in[4];
    const float* t_edge  = (const float*)d_in[5];
    const float* t_node  = (const float*)d_in[6];
    const float* eW1 = (const float*)d_in[7];
    const float* eb1 = (const float*)d_in[8];
    const float* eW2 = (const float*)d_in[9];
    const float* eb2 = (const float*)d_in[10];
    const float* cW1 = (const float*)d_in[11];
    const float* cb1 = (const float*)d_in[12];
    const float* cW2 = (const float*)d_in[13];
    const float* nW1 = (const float*)d_in[14];
    const float* nb1 = (const float*)d_in[15];
    const float* nW2 = (const float*)d_in[16];
    const float* nb2 = (const float*)d_in[17];

    float* h_out     = (float*)d_out;                    // [N_NODES, D]
    float* coord_out = (float*)d_out + N_NODES * D;      // [N_NODES, 3]

    // Workspace layout
    float* ws    = (float*)d_ws;
    float* m_i   = ws;                         ws += N_NODES * D;     // 1,280,000
    float* eW1t  = ws;                         ws += D * K_EDGE_PAD;  //    41,472
    float* eW2t  = ws;                         ws += D * D;           //    16,384
    float* cW1t  = ws;                         ws += D * D;
    float* nW1t  = ws;                         ws += D * K_NODE;      //    40,960
    float* nW2t  = ws;                         ws += D * D;

    // Weight prep (padded, N-major transposed for contiguous b64 B-fragments)
    egnn_transpose_pad<<<(D * K_EDGE_PAD + 255) / 256, 256, 0, stream>>>(eW1, eW1t, K_EDGE, K_EDGE_PAD);
    egnn_transpose_pad<<<(D * D + 255) / 256, 256, 0, stream>>>(eW2, eW2t, D, D);
    egnn_transpose_pad<<<(D * D + 255) / 256, 256, 0, stream>>>(cW1, cW1t, D, D);
    egnn_transpose_pad<<<(D * K_NODE + 255) / 256, 256, 0, stream>>>(nW1, nW1t, K_NODE, K_NODE);
    egnn_transpose_pad<<<(D * D + 255) / 256, 256, 0, stream>>>(nW2, nW2t, D, D);

    egnn_zero_kernel<<<(N_NODES * D + 255) / 256, 256, 0, stream>>>(m_i, coord_out);

    egnn_edge_kernel<<<N_EDGES / 32, 64, 0, stream>>>(
        h, diff, dist_sq, esrc, edst, t_edge,
        eW1t, eb1, eW2t, eb2, cW1t, cb1, cW2, m_i, coord_out);

    egnn_node_kernel<<<(N_NODES + 31) / 32, 64, 0, stream>>>(
        h, m_i, t_node, nW1t, nb1, nW2t, nb2, h_out);
}